// LessonGCN_44702019616965
// MI455X (gfx1250) — compile-verified
//
#include <hip/hip_runtime.h>

#define IN_DIM 64
#define HID_DIM 128

typedef float v2f __attribute__((ext_vector_type(2)));
typedef float v8f __attribute__((ext_vector_type(8)));

// ---------------- degree / normalization ----------------

__global__ void k_init_deg(float* __restrict__ deg, int n) {
    int i = blockIdx.x * blockDim.x + threadIdx.x;
    if (i < n) deg[i] = 1.0f;  // +1 self loop
}

__global__ void k_deg_accum(const int* __restrict__ dst, float* __restrict__ deg, int nE) {
    int e = blockIdx.x * blockDim.x + threadIdx.x;
    if (e < nE) atomicAdd(&deg[dst[e]], 1.0f);
}

__global__ void k_rsqrt(float* __restrict__ deg, int n) {
    int i = blockIdx.x * blockDim.x + threadIdx.x;
    if (i < n) deg[i] = rsqrtf(deg[i]);  // deg -> dinv in place
}

// ---------------- layer-1 GEMM: h1 = x @ W1 (fp32 WMMA 16x16x4) ----------------
// One wave per 16x16 output tile; block of 8 waves covers all 128 output cols
// for one 16-row stripe. x tile staged in LDS.
// A layout (32-bit 16x4, ISA 7.12.2): lanes 0-15 row M=lane, v0=K0,v1=K1;
// lanes 16-31 row M=lane-16, v0=K2,v1=K3. B (4x16) mirrored with N across lanes.
__global__ void k_gemm1(const float* __restrict__ x, const float* __restrict__ W1,
                        float* __restrict__ h1, int nNodes) {
    __shared__ float xs[16 * IN_DIM];
    const int rowBase = blockIdx.x * 16;
    const int wave = threadIdx.x >> 5;
    const int lane = threadIdx.x & 31;
    const int half = lane >> 4;   // 0: K lo pair, 1: K hi pair
    const int l16  = lane & 15;
    const int colBase = wave * 16;

    // stage 16x64 fp32 tile of x into LDS: 256 threads x 4 floats
    for (int i = threadIdx.x * 4; i < 16 * IN_DIM; i += 256 * 4) {
        int r = i / IN_DIM;
        int k = i % IN_DIM;
        int gr = rowBase + r;
        float4 v;
        if (gr < nNodes) v = *(const float4*)(x + (size_t)gr * IN_DIM + k);
        else             v = make_float4(0.f, 0.f, 0.f, 0.f);
        *(float4*)(xs + i) = v;
    }
    __syncthreads();

    v8f c = {};
#pragma unroll
    for (int kb = 0; kb < IN_DIM / 4; ++kb) {
        const int kA = kb * 4 + half * 2;
        v2f a, b;
        a.x = xs[l16 * IN_DIM + kA];
        a.y = xs[l16 * IN_DIM + kA + 1];
        b.x = W1[(size_t)kA       * HID_DIM + colBase + l16];
        b.y = W1[(size_t)(kA + 1) * HID_DIM + colBase + l16];
        c = __builtin_amdgcn_wmma_f32_16x16x4_f32(
                /*neg_a=*/false, a, /*neg_b=*/false, b,
                /*c_mod=*/(short)0, c, /*reuse_a=*/false, /*reuse_b=*/false);
    }

    // D layout: vgpr v -> row M = v + 8*half, col N = l16
#pragma unroll
    for (int v = 0; v < 8; ++v) {
        int gr = rowBase + v + 8 * half;
        if (gr < nNodes) h1[(size_t)gr * HID_DIM + colBase + l16] = c[v];
    }
}

// ---------------- self-loop contribution: agg1 = h1 * dinv_i^2 ----------------
__global__ void k_selfloop1(const float* __restrict__ h1, const float* __restrict__ dinv,
                            float* __restrict__ agg1, int total) {
    int i = blockIdx.x * blockDim.x + threadIdx.x;
    if (i >= total) return;
    float di = dinv[i >> 7];  // HID_DIM == 128
    agg1[i] = h1[i] * (di * di);
}

// ---------------- layer-1 edge scatter: 32 lanes per edge, float4 each ----------------
__global__ void k_edge1(const int* __restrict__ src, const int* __restrict__ dst,
                        const float* __restrict__ h1, const float* __restrict__ dinv,
                        float* __restrict__ agg1, int nE) {
    int gid = blockIdx.x * blockDim.x + threadIdx.x;
    int e = gid >> 5;
    if (e >= nE) return;
    int f = (gid & 31) * 4;
    int s = src[e], d = dst[e];
    float nrm = dinv[s] * dinv[d];
    const float4 hv = *(const float4*)(h1 + (size_t)s * HID_DIM + f);
    float* o = agg1 + (size_t)d * HID_DIM + f;
    atomicAdd(o + 0, hv.x * nrm);
    atomicAdd(o + 1, hv.y * nrm);
    atomicAdd(o + 2, hv.z * nrm);
    atomicAdd(o + 3, hv.w * nrm);
}

// ---------------- layer-2: s = relu(agg1 + b1) @ W2 ; out = b2 + s*dinv^2 ----------------
__global__ void k_relu_gemv2(const float* __restrict__ agg1, const float* __restrict__ b1,
                             const float* __restrict__ W2, const float* __restrict__ b2,
                             const float* __restrict__ dinv, float* __restrict__ sbuf,
                             float* __restrict__ out, int n) {
    int node = blockIdx.x * (blockDim.x >> 5) + (threadIdx.x >> 5);
    int lane = threadIdx.x & 31;
    if (node >= n) return;
    float acc = 0.f;
#pragma unroll
    for (int j = 0; j < 4; ++j) {
        int f = lane + j * 32;
        float hv = agg1[(size_t)node * HID_DIM + f] + b1[f];
        hv = fmaxf(hv, 0.f);
        acc = fmaf(hv, W2[f], acc);
    }
#pragma unroll
    for (int off = 16; off; off >>= 1) acc += __shfl_xor(acc, off, 32);
    if (lane == 0) {
        float di = dinv[node];
        sbuf[node] = acc;
        out[node] = b2[0] + acc * di * di;
    }
}

// ---------------- layer-2 edge scatter (scalar per edge) ----------------
__global__ void k_edge2(const int* __restrict__ src, const int* __restrict__ dst,
                        const float* __restrict__ sbuf, const float* __restrict__ dinv,
                        float* __restrict__ out, int nE) {
    int e = blockIdx.x * blockDim.x + threadIdx.x;
    if (e >= nE) return;
    int a = src[e], b = dst[e];
    atomicAdd(&out[b], sbuf[a] * dinv[a] * dinv[b]);
}

// ---------------- launcher ----------------
extern "C" void kernel_launch(void* const* d_in, const int* in_sizes, int n_in,
                              void* d_out, int out_size, void* d_ws, size_t ws_size,
                              hipStream_t stream) {
    const float* x  = (const float*)d_in[0];
    const int*   ei = (const int*)d_in[1];
    const float* W1 = (const float*)d_in[2];
    const float* b1 = (const float*)d_in[3];
    const float* W2 = (const float*)d_in[4];
    const float* b2 = (const float*)d_in[5];
    float* out = (float*)d_out;

    const int nNodes = in_sizes[0] / IN_DIM;
    const int nE     = in_sizes[1] / 2;
    const int* src = ei;
    const int* dst = ei + nE;

    // workspace layout (floats): dinv[N] | h1[N*128] | agg1[N*128] | s[N]
    float* ws   = (float*)d_ws;
    float* dinv = ws;
    float* h1   = dinv + nNodes;
    float* agg1 = h1 + (size_t)nNodes * HID_DIM;
    float* sbuf = agg1 + (size_t)nNodes * HID_DIM;

    const int T = 256;
    k_init_deg <<<(nNodes + T - 1) / T, T, 0, stream>>>(dinv, nNodes);
    k_deg_accum<<<(nE + T - 1) / T, T, 0, stream>>>(dst, dinv, nE);
    k_rsqrt    <<<(nNodes + T - 1) / T, T, 0, stream>>>(dinv, nNodes);

    k_gemm1<<<(nNodes + 15) / 16, 256, 0, stream>>>(x, W1, h1, nNodes);

    {
        int tot = nNodes * HID_DIM;
        k_selfloop1<<<(tot + T - 1) / T, T, 0, stream>>>(h1, dinv, agg1, tot);
    }
    {
        long long work = (long long)nE * 32;
        k_edge1<<<(int)((work + T - 1) / T), T, 0, stream>>>(src, dst, h1, dinv, agg1, nE);
    }

    k_relu_gemv2<<<(nNodes + 7) / 8, 256, 0, stream>>>(agg1, b1, W2, b2, dinv, sbuf, out, nNodes);
    k_edge2<<<(nE + T - 1) / T, T, 0, stream>>>(src, dst, sbuf, dinv, out, nE);
}